// MSM_15556371546959
// MI455X (gfx1250) — compile-verified
//
#include <hip/hip_runtime.h>
#include <cstdint>

// ---------------------------------------------------------------------------
// MSM forward for MI455X (gfx1250).
//  Kernel 1: fused integral images (IX=cumsum_H, IY=cumsum_W, IT=cumsum_W(IX))
//            TDM double-buffered tile streaming + wave32 shuffle scans.
//  Kernel 2: whole network, 1 workgroup per batch, activations in LDS,
//            all linears via V_WMMA_F32_16X16X4_F32 (fp32-exact).
// ---------------------------------------------------------------------------

typedef float v2f __attribute__((ext_vector_type(2)));
typedef float v8f __attribute__((ext_vector_type(8)));
typedef unsigned int v4u __attribute__((ext_vector_type(4)));
typedef int v4i __attribute__((ext_vector_type(4)));
typedef int v8i __attribute__((ext_vector_type(8)));

#define IMG_HW   512
#define IMG_PLN  (IMG_HW * IMG_HW)      // 262144
#define NPLANES  48                     // B(16) * C(3)
#define NBATCH   16
#define NCLS     20

// ----- model shape tables (derived from the reference) ---------------------
constexpr int UPB_N0[4] = {0, 1, 4, 12};
constexpr int UPB_N1[4] = {1, 2, 4, 8};
constexpr int U_CH[4]   = {64, 32, 16, 8};   // UNFOLD_CH == FUSE_CH
constexpr int U_G[4]    = {16, 8, 4, 2};     // UNFOLD_G  == FUSE_G
constexpr int U_IN[4]   = {128, 64, 32, 16}; // "up" per layer
constexpr int M_Y[4]    = {3, 8, 20, 48};    // unfold output tokens
constexpr int F_OUT[4]  = {128, 64, 32, 16}; // fuse out_ch
constexpr int XS_OFF[4] = {0, 192, 448, 768};

// ----- parameter table (jax sorted-key tree flatten of `params`) -----------
// p[0..63]  : fuse[i]  {mlp1 W,b, mlp2 W,b, n1 g,b, n2 g,b, pk W,b, pq W,b, pv W,b, sp W,b}
// p[64,65]  : head W,b
// p[66,67]  : mp W,b
// p[68]     : query (1,1,128)
// p[69..124]: unfold[i] {inter0 W,b, inter1 W,b, mlp1 W,b, mlp2 W,b, n1 g,b, n2 g,b, pr W,b}
struct Params { const float* p[125]; };
__host__ __device__ __forceinline__ constexpr int FS(int li, int k) { return 16 * li + k; }
__host__ __device__ __forceinline__ constexpr int UF(int li, int k) { return 69 + 14 * li + k; }
#define IDX_HEAD_W 64
#define IDX_HEAD_B 65
#define IDX_MP_W   66
#define IDX_MP_B   67
#define IDX_QUERY  68

// ----- small math helpers ---------------------------------------------------
__device__ __forceinline__ float sigmoid_f(float x) { return 1.0f / (1.0f + __expf(-x)); }
__device__ __forceinline__ float gelu_f(float x) {
  return 0.5f * x * (1.0f + erff(x * 0.70710678118654752f)); // exact gelu
}

// ---------------------------------------------------------------------------
// TDM: hand-packed Tensor DMA descriptor, 2-D fp32 tile -> LDS.
// clang-23 / therock arity: 6 args (g0 v4u, g1 v8i, g2 v4i, g3 v4i, v8i, cpol)
// ---------------------------------------------------------------------------
#if defined(__has_builtin)
#if __has_builtin(__builtin_amdgcn_tensor_load_to_lds) && __has_builtin(__builtin_amdgcn_s_wait_tensorcnt)
#define MSM_USE_TDM 1
#endif
#endif
#ifndef MSM_USE_TDM
#define MSM_USE_TDM 0
#endif

#if MSM_USE_TDM
__device__ __forceinline__ void tdm_load_2d_f32(unsigned lds_off, const float* gsrc,
                                                int rows, int cols, int src_stride_elems) {
  unsigned long long ga = (unsigned long long)(uintptr_t)gsrc;
  v4u g0;
  g0[0] = 1u;                                          // count=1 valid descriptor
  g0[1] = lds_off;                                     // D#[63:32]  lds_addr (bytes)
  g0[2] = (unsigned)(ga & 0xffffffffu);                // D#[95:64]  global_addr lo
  g0[3] = (unsigned)((ga >> 32) & 0x01ffffffu)         // D#[120:96] global_addr hi
        | (2u << 30);                                  // D#[127:126] type = 2 (image)
  v8i g1;
  g1[0] = (2 << 16);                                   // data_size=2 (4B); wg_mask=0
  g1[1] = (int)(((unsigned)cols & 0xffffu) << 16);     // tensor_dim0 [63:48]
  g1[2] = (int)((((unsigned)cols >> 16) & 0xffffu)     // tensor_dim0 [79:64]
        | (((unsigned)rows & 0xffffu) << 16));         // tensor_dim1 [95:80]
  g1[3] = (int)((((unsigned)rows >> 16) & 0xffffu)     // tensor_dim1 [111:96]
        | (((unsigned)cols & 0xffffu) << 16));         // tile_dim0   [127:112]
  g1[4] = (int)((unsigned)rows & 0xffffu);             // tile_dim1; tile_dim2=0
  g1[5] = src_stride_elems;                            // tensor_dim0_stride lo32
  g1[6] = 0;
  g1[7] = 0;
  v4i z4 = {0, 0, 0, 0};
  v8i z8 = {0, 0, 0, 0, 0, 0, 0, 0};
  __builtin_amdgcn_tensor_load_to_lds(g0, g1, z4, z4, z8, 0);
}
#endif

// ---------------------------------------------------------------------------
// Block-wide (512 thread, 16 wave) inclusive scan, wave32.
// ---------------------------------------------------------------------------
__device__ __forceinline__ float block_scan512(float v, float* wpart) {
  const int lane = threadIdx.x & 31;
  const int wv   = threadIdx.x >> 5;
  float s = v;
#pragma unroll
  for (int d = 1; d < 32; d <<= 1) {
    float t = __shfl_up(s, (unsigned)d, 32);
    if (lane >= d) s += t;
  }
  __syncthreads();                    // protect wpart reuse across calls
  if (lane == 31) wpart[wv] = s;
  __syncthreads();
  if (wv == 0) {
    float p = (lane < 16) ? wpart[lane] : 0.0f;
#pragma unroll
    for (int d = 1; d < 16; d <<= 1) {
      float t = __shfl_up(p, (unsigned)d, 32);
      if (lane >= d) p += t;
    }
    if (lane < 16) wpart[lane] = p;
  }
  __syncthreads();
  float off = (wv > 0) ? wpart[wv - 1] : 0.0f;
  return s + off;
}

// ---------------------------------------------------------------------------
// Kernel 1: fused integral images. One workgroup per (b,c) plane; thread =
// column. Single pass: 1x read of I, 1x write each of IX/IY/IT (traffic floor).
// ---------------------------------------------------------------------------
__global__ __launch_bounds__(512) void integral_kernel(
    const float* __restrict__ I, float* __restrict__ IX,
    float* __restrict__ IY, float* __restrict__ IT) {
  __shared__ float tile[2][8 * IMG_HW];   // 2 x 16KB double buffer
  __shared__ float wpart[16];

  const int plane = blockIdx.x;
  const int w = threadIdx.x;
  const int wv = threadIdx.x >> 5;
  const size_t base = (size_t)plane * IMG_PLN;
  const float* Ip = I + base;
  float* IXp = IX + base;
  float* IYp = IY + base;
  float* ITp = IT + base;

  const int R = 8, NT = IMG_HW / R;

#if MSM_USE_TDM
  if (wv == 0) {
    tdm_load_2d_f32((unsigned)(uintptr_t)(void*)&tile[0][0], Ip, R, IMG_HW, IMG_HW);
    __builtin_amdgcn_s_wait_tensorcnt(0);
  }
  __syncthreads();
#else
  for (int idx = threadIdx.x; idx < R * IMG_HW; idx += blockDim.x) tile[0][idx] = Ip[idx];
  __syncthreads();
#endif

  float carry = 0.0f;   // H-direction running sum for this thread's column
  for (int tb = 0; tb < NT; ++tb) {
    const int cur = tb & 1;
#if MSM_USE_TDM
    if (wv == 0 && tb + 1 < NT)       // prefetch next tile while scanning this one
      tdm_load_2d_f32((unsigned)(uintptr_t)(void*)&tile[cur ^ 1][0],
                      Ip + (size_t)(tb + 1) * R * IMG_HW, R, IMG_HW, IMG_HW);
#endif
    for (int r = 0; r < R; ++r) {
      const int h = tb * R + r;
      float v = tile[cur][r * IMG_HW + w];
      carry += v;                                    // IX[h][w]
      float iy = block_scan512(v, wpart);            // cumsum along W of I
      float it = block_scan512(carry, wpart);        // cumsum along W of IX
      IXp[h * IMG_HW + w] = carry;
      IYp[h * IMG_HW + w] = iy;
      ITp[h * IMG_HW + w] = it;
    }
#if MSM_USE_TDM
    if (wv == 0) __builtin_amdgcn_s_wait_tensorcnt(0);
    __syncthreads();
#else
    __syncthreads();
    if (tb + 1 < NT)
      for (int idx = threadIdx.x; idx < R * IMG_HW; idx += blockDim.x)
        tile[cur ^ 1][idx] = Ip[(size_t)(tb + 1) * R * IMG_HW + idx];
    __syncthreads();
#endif
  }
}

// ---------------------------------------------------------------------------
// Generic fp32 WMMA linear:  Y[m][n] = act( sum_k A[m][k]*W[k][n] + bias[n] )
// 16x16 output tiles over the block's 8 waves; V_WMMA_F32_16X16X4_F32.
//   A fragment : lane l holds (m = l&15, k = k0 + 2*(l>>4) + vgpr)
//   B fragment : lane l holds (n = l&15, k = k0 + 2*(l>>4) + vgpr)
//   C/D        : lane l, vgpr v -> (m = v + 8*(l>>4), n = l&15)
// ---------------------------------------------------------------------------
#define NET_NWAVES 8
__device__ void lin_wmma(const float* __restrict__ A, int lda, int M, int K,
                         const float* __restrict__ W, const float* __restrict__ bias,
                         int N, float* __restrict__ Y, int ldy, int act /*0=none,1=gelu*/) {
  const int wave = threadIdx.x >> 5;
  const int lane = threadIdx.x & 31;
  const int lm = lane & 15;
  const int hi = lane >> 4;
  const int tilesM = (M + 15) >> 4;
  const int tilesN = (N + 15) >> 4;
  for (int t = wave; t < tilesM * tilesN; t += NET_NWAVES) {
    const int tm = (t / tilesN) << 4;
    const int tn = (t % tilesN) << 4;
    const int m = tm + lm;
    const int n = tn + lm;
    v8f acc = {0.f, 0.f, 0.f, 0.f, 0.f, 0.f, 0.f, 0.f};
    for (int k0 = 0; k0 < K; k0 += 4) {
      const int ka = k0 + (hi << 1);
      v2f a, b;
      a.x = (m < M && ka     < K) ? A[m * lda + ka    ] : 0.0f;
      a.y = (m < M && ka + 1 < K) ? A[m * lda + ka + 1] : 0.0f;
      b.x = (n < N && ka     < K) ? W[ka       * N + n] : 0.0f;
      b.y = (n < N && ka + 1 < K) ? W[(ka + 1) * N + n] : 0.0f;
      acc = __builtin_amdgcn_wmma_f32_16x16x4_f32(false, a, false, b,
                                                  (short)0, acc, false, false);
    }
#pragma unroll
    for (int v = 0; v < 8; ++v) {
      const int mm = tm + v + (hi << 3);
      if (mm < M && n < N) {
        float val = acc[v] + bias[n];
        if (act == 1) val = gelu_f(val);
        Y[mm * ldy + n] = val;
      }
    }
  }
}

// per-token LayerNorm over C channels (tiny M -> thread per token)
__device__ void ln_tokens(float* X, int M, int C, const float* g, const float* b) {
  for (int m = threadIdx.x; m < M; m += blockDim.x) {
    float mean = 0.f;
    for (int c = 0; c < C; ++c) mean += X[m * C + c];
    mean /= (float)C;
    float var = 0.f;
    for (int c = 0; c < C; ++c) { float d = X[m * C + c] - mean; var += d * d; }
    var /= (float)C;
    float r = rsqrtf(var + 1e-5f);
    for (int c = 0; c < C; ++c) X[m * C + c] = (X[m * C + c] - mean) * r * g[c] + b[c];
  }
}

// ----- integral-image sampling (reference sample_rects_topleft) -------------
__device__ __forceinline__ float ii_get(const float* A, int plane, int xi, int yi) {
  xi &= (IMG_HW - 1); yi &= (IMG_HW - 1);   // JAX negative-index wrap (only -1 occurs)
  return A[(size_t)plane * IMG_PLN + xi * IMG_HW + yi];
}

__device__ float sample_tl(const float* I, const float* IX, const float* IY,
                           const float* IT, int plane, float x, float y) {
  int X = (int)ceilf(x), Y = (int)ceilf(y);
  float dx = (float)X - x, dy = (float)Y - y;
  float wx1 = 0.5f * dx * dx, wx2 = dx - wx1;
  float wy1 = 0.5f * dy * dy, wy2 = dy - wy1;
  float s_a = ii_get(IT, plane, X, Y);
  float s_e1 =
      wy1 * (ii_get(IX, plane, X, Y - 1) - 0.5f * ii_get(I, plane, X, Y - 1) - 0.5f * ii_get(I, plane, 0, Y - 1)) +
      wy2 * (ii_get(IX, plane, X, Y)     - 0.5f * ii_get(I, plane, X, Y)     - 0.5f * ii_get(I, plane, 0, Y));
  float s_e2 =
      wx1 * (ii_get(IY, plane, X - 1, Y) - 0.5f * ii_get(I, plane, X - 1, Y) - 0.5f * ii_get(I, plane, X - 1, 0)) +
      wx2 * (ii_get(IY, plane, X, Y)     - 0.5f * ii_get(I, plane, X, Y)     - 0.5f * ii_get(I, plane, X, 0));
  float s_c = wx1 * wy1 * ii_get(I, plane, X - 1, Y - 1) + wx2 * wy1 * ii_get(I, plane, X, Y - 1) +
              wx1 * wy2 * ii_get(I, plane, X - 1, Y)     + wx2 * wy2 * ii_get(I, plane, X, Y);
  return s_a - s_e1 - s_e2 + s_c;
}

// ---------------------------------------------------------------------------
// Unfold layer (tokens: n0+n1 in, M=2n0+3n1 out)
// ---------------------------------------------------------------------------
__device__ void do_unfold(int li, int b, const Params& P,
                          const float* I, const float* IX, const float* IY, const float* IT,
                          const float* x_in, float* x_out,
                          float* lob, float* yb, float* srb, float* rb, float* mb,
                          float* zb, float* hb, float* z2b) {
  const int n0 = UPB_N0[li], n1 = UPB_N1[li];
  const int up = U_IN[li];
  const int down = U_CH[li], g = U_G[li];
  const int ic = 12 + 3 * g;
  const int M = 2 * n0 + 3 * n1;
  const int nthr = blockDim.x;

  // 1) inter projections, then permute (n, u*ic+c) -> token-major y
  for (int i = 0; i < 2; ++i) {
    const int ni = (i == 0) ? n0 : n1;
    if (ni == 0) continue;
    const int ti = (i == 0) ? 2 : 3;
    const int N = ic * ti;
    const float* A = x_in + ((i == 0) ? 0 : n0) * up;
    lin_wmma(A, up, ni, up, P.p[UF(li, i * 2)], P.p[UF(li, i * 2 + 1)], N, lob, N, 0);
    __syncthreads();
    const int base = (i == 0) ? 0 : 2 * n0;
    for (int idx = threadIdx.x; idx < ni * ti * ic; idx += nthr) {
      int c = idx % ic, r = idx / ic;
      int n = r / ti, u = r % ti;
      yb[(base + r) * ic + c] = lob[n * N + u * ic + c];
    }
    __syncthreads();
  }

  // 2) rect sampling from integral images (4-corner inclusion/exclusion)
  for (int idx = threadIdx.x; idx < M * 3; idx += nthr) {
    int m = idx / 3, r = idx % 3;
    const float* pp = &yb[m * ic + r * 4];
    float x1 = sigmoid_f(pp[0]) * 511.0f;
    float y1 = sigmoid_f(pp[1]) * 511.0f;
    float x2 = sigmoid_f(pp[2]) * 511.0f;
    float y2 = sigmoid_f(pp[3]) * 511.0f;
    for (int c = 0; c < 3; ++c) {
      int plane = b * 3 + c;
      float S = sample_tl(I, IX, IY, IT, plane, x2, y2)
              - sample_tl(I, IX, IY, IT, plane, x1, y2)
              - sample_tl(I, IX, IY, IT, plane, x2, y1)
              + sample_tl(I, IX, IY, IT, plane, x1, y1);
      srb[idx * 3 + c] = S;
    }
  }
  __syncthreads();

  // 3) rects = gelu(rects @ pr + b)   (M*3 x 3) @ (3 x down)
  lin_wmma(srb, 3, M * 3, 3, P.p[UF(li, 12)], P.p[UF(li, 13)], down, rb, down, 1);

  // 4) softmax over the NR=3 axis of the mask channels (independent of 3))
  for (int idx = threadIdx.x; idx < M * g; idx += nthr) {
    int m = idx / g, gi = idx % g;
    float e0 = yb[m * ic + 12 + 0 * g + gi];
    float e1 = yb[m * ic + 12 + 1 * g + gi];
    float e2 = yb[m * ic + 12 + 2 * g + gi];
    float mx = fmaxf(e0, fmaxf(e1, e2));
    e0 = __expf(e0 - mx); e1 = __expf(e1 - mx); e2 = __expf(e2 - mx);
    float inv = 1.0f / (e0 + e1 + e2);
    mb[(m * 3 + 0) * g + gi] = e0 * inv;
    mb[(m * 3 + 1) * g + gi] = e1 * inv;
    mb[(m * 3 + 2) * g + gi] = e2 * inv;
  }
  __syncthreads();

  // 5) einsum bmsg,bmsgc->bmgc
  const int dg = down / g;
  for (int idx = threadIdx.x; idx < M * down; idx += nthr) {
    int m = idx / down, ch = idx % down;
    int gi = ch / dg;
    zb[idx] = mb[(m * 3 + 0) * g + gi] * rb[(m * 3 + 0) * down + ch]
            + mb[(m * 3 + 1) * g + gi] * rb[(m * 3 + 1) * down + ch]
            + mb[(m * 3 + 2) * g + gi] * rb[(m * 3 + 2) * down + ch];
  }
  __syncthreads();

  // 6) LN -> MLP -> LN + residual
  ln_tokens(zb, M, down, P.p[UF(li, 8)], P.p[UF(li, 9)]);
  __syncthreads();
  lin_wmma(zb, down, M, down, P.p[UF(li, 4)], P.p[UF(li, 5)], 4 * down, hb, 4 * down, 1);
  __syncthreads();
  lin_wmma(hb, 4 * down, M, 4 * down, P.p[UF(li, 6)], P.p[UF(li, 7)], down, z2b, down, 0);
  __syncthreads();
  ln_tokens(z2b, M, down, P.p[UF(li, 10)], P.p[UF(li, 11)]);
  __syncthreads();
  for (int idx = threadIdx.x; idx < M * down; idx += nthr) x_out[idx] = z2b[idx] + zb[idx];
  __syncthreads();
}

// ---------------------------------------------------------------------------
// Fuse layer (grouped cross-attention + MLP), u: M_Y[li] tokens -> n0+n1 tokens
// ---------------------------------------------------------------------------
__device__ void do_fuse(int li, const Params& P, const float* x_in, const float* u_in,
                        float* u_out, float* Kb, float* Qb, float* Vb, float* scb,
                        float* zb, float* hb, float* z2b) {
  const int n0 = UPB_N0[li], n1 = UPB_N1[li];
  const int ntok = n0 + n1;
  const int Mu = M_Y[li];
  const int down = U_CH[li], g = U_G[li];
  const int up = (li == 0) ? 128 : U_CH[li - 1];
  const int oc = F_OUT[li];
  const int nthr = blockDim.x;

  // independent projections -> disjoint buffers (no syncs needed between)
  lin_wmma(x_in, up, ntok, up, P.p[FS(li, 14)], P.p[FS(li, 15)], oc, scb, oc, 0);   // shortcut
  lin_wmma(u_in, down, Mu, down, P.p[FS(li, 8)], P.p[FS(li, 9)], down, Kb, down, 0); // K
  lin_wmma(x_in, up, ntok, up, P.p[FS(li, 10)], P.p[FS(li, 11)], down, Qb, down, 0); // Q
  lin_wmma(u_in, down, Mu, down, P.p[FS(li, 12)], P.p[FS(li, 13)], oc, Vb, oc, 0);   // V
  __syncthreads();

  const int dg = down / g, dgo = oc / g;
  const float scale = rsqrtf((float)down);
  for (int idx = threadIdx.x; idx < ntok * g; idx += nthr) {
    int n = idx / g, gi = idx % g;
    int tcnt, ub;
    if (n < n0) { tcnt = 2; ub = n * 2; }
    else        { tcnt = 3; ub = 2 * n0 + (n - n0) * 3; }
    float e[3];
    float mx = -1e30f;
    for (int t = 0; t < tcnt; ++t) {
      float d = 0.f;
      for (int cc = 0; cc < dg; ++cc)
        d += Kb[(ub + t) * down + gi * dg + cc] * Qb[n * down + gi * dg + cc];
      e[t] = d * scale;
      mx = fmaxf(mx, e[t]);
    }
    float ssum = 0.f;
    for (int t = 0; t < tcnt; ++t) { e[t] = __expf(e[t] - mx); ssum += e[t]; }
    float inv = 1.0f / ssum;
    for (int cc = 0; cc < dgo; ++cc) {
      float a = 0.f;
      for (int t = 0; t < tcnt; ++t) a += e[t] * inv * Vb[(ub + t) * oc + gi * dgo + cc];
      zb[n * oc + gi * dgo + cc] = a;
    }
  }
  __syncthreads();

  ln_tokens(zb, ntok, oc, P.p[FS(li, 4)], P.p[FS(li, 5)]);
  __syncthreads();
  lin_wmma(zb, oc, ntok, oc, P.p[FS(li, 0)], P.p[FS(li, 1)], 4 * oc, hb, 4 * oc, 1);
  __syncthreads();
  lin_wmma(hb, 4 * oc, ntok, 4 * oc, P.p[FS(li, 2)], P.p[FS(li, 3)], oc, z2b, oc, 0);
  __syncthreads();
  ln_tokens(z2b, ntok, oc, P.p[FS(li, 6)], P.p[FS(li, 7)]);
  __syncthreads();
  for (int idx = threadIdx.x; idx < ntok * oc; idx += nthr) u_out[idx] = z2b[idx] + scb[idx];
  __syncthreads();
}

// ---------------------------------------------------------------------------
// Kernel 2: whole network. grid = 16 (one WG per batch), block = 256 (8 waves)
// ---------------------------------------------------------------------------
__global__ __launch_bounds__(256) void msm_network_kernel(
    const float* __restrict__ I, const float* __restrict__ IX,
    const float* __restrict__ IY, const float* __restrict__ IT,
    Params P, float* __restrict__ out) {
  __shared__ float qbuf[128];
  __shared__ float xsbuf[1152];   // xs[0..3], offsets XS_OFF
  __shared__ float lob[512];      // inter lin scratch
  __shared__ float yb[896];       // permuted y (M x ic)
  __shared__ float srb[448];      // sampled rects (M*3 x 3)
  __shared__ float rb[1152];      // rects proj (M*3 x down)
  __shared__ float mb[288];       // softmax masks
  __shared__ float zb[384];
  __shared__ float hb[1536];      // mlp hidden
  __shared__ float z2b[384];
  __shared__ float scb[384];      // fuse shortcut
  __shared__ float Kb[384];
  __shared__ float Qb[192];
  __shared__ float Vb[768];
  __shared__ float ubufA[384];
  __shared__ float ubufB[384];

  const int b = blockIdx.x;

  for (int idx = threadIdx.x; idx < 128; idx += blockDim.x) qbuf[idx] = P.p[IDX_QUERY][idx];
  __syncthreads();

  // unfold stack
  for (int li = 0; li < 4; ++li) {
    const float* xin = (li == 0) ? qbuf : (xsbuf + XS_OFF[li - 1]);
    do_unfold(li, b, P, I, IX, IY, IT, xin, xsbuf + XS_OFF[li],
              lob, yb, srb, rb, mb, zb, hb, z2b);
  }

  // u = lin(gelu(xs3), mp)
  for (int idx = threadIdx.x; idx < 48 * 8; idx += blockDim.x)
    hb[idx] = gelu_f(xsbuf[XS_OFF[3] + idx]);
  __syncthreads();
  lin_wmma(hb, 8, 48, 8, P.p[IDX_MP_W], P.p[IDX_MP_B], 8, ubufA, 8, 0);
  __syncthreads();

  // fuse stack (3 -> 0)
  float* ucur = ubufA;
  float* unxt = ubufB;
  for (int li = 3; li >= 0; --li) {
    const float* xin = (li == 0) ? qbuf : (xsbuf + XS_OFF[li - 1]);
    do_fuse(li, P, xin, ucur, unxt, Kb, Qb, Vb, scb, zb, hb, z2b);
    float* t = ucur; ucur = unxt; unxt = t;
  }

  // head: silu(u0) @ W + b   (1 x 128) @ (128 x 20)
  const float* headW = P.p[IDX_HEAD_W];
  const float* headB = P.p[IDX_HEAD_B];
  for (int cls = threadIdx.x; cls < NCLS; cls += blockDim.x) {
    float acc = headB[cls];
    for (int c = 0; c < 128; ++c) {
      float s = ucur[c];
      acc += (s / (1.0f + __expf(-s))) * headW[c * NCLS + cls];
    }
    out[b * NCLS + cls] = acc;
  }
}

// ---------------------------------------------------------------------------
extern "C" void kernel_launch(void* const* d_in, const int* in_sizes, int n_in,
                              void* d_out, int out_size, void* d_ws, size_t ws_size,
                              hipStream_t stream) {
  (void)in_sizes; (void)out_size; (void)ws_size;
  const float* I = (const float*)d_in[0];

  // workspace: IX, IY, IT (3 x 48MB) in d_ws
  const size_t planeElems = (size_t)NPLANES * IMG_PLN;
  float* IX = (float*)d_ws;
  float* IY = IX + planeElems;
  float* IT = IY + planeElems;

  integral_kernel<<<NPLANES, 512, 0, stream>>>(I, IX, IY, IT);

  Params P;
  const int np = (n_in - 1 < 125) ? (n_in - 1) : 125;
  for (int j = 0; j < 125; ++j) P.p[j] = (j < np) ? (const float*)d_in[1 + j] : (const float*)d_in[0];

  msm_network_kernel<<<NBATCH, 256, 0, stream>>>(I, IX, IY, IT, P, (float*)d_out);
}